// PC_DONN_88450556494249
// MI455X (gfx1250) — compile-verified
//
#include <hip/hip_runtime.h>

// ---------------------------------------------------------------------------
// PC-DONN forward, analytically collapsed.
//
// |H(fx,fy)| == 1 for every sampled frequency (cutoff 1/WL >> Nyquist) and the
// layer phase masks are unit modulus, so every propagation step is unitary and
// the plane-summed intensity reduces exactly to
//     out[b] = (1/M) sum_m sum_n x[b,n]^2 * |T_m[n]|^2,
// T_m = fftshift(E F_m E),  E[a,b] = exp(+2*pi*i*a*b/N)  (N^2 factors cancel).
// The inverse 2-D DFT is evaluated as complex GEMMs on the FP32 WMMA unit.
// ---------------------------------------------------------------------------

#define NN   128
#define NPIX (NN * NN)
#define FSTR 129   // LDS row stride for F (float2 units) -> conflict-light
#define PSTR 17    // LDS row stride for P (float2 units)

typedef float v2f __attribute__((ext_vector_type(2)));
typedef float v8f __attribute__((ext_vector_type(8)));

__device__ __forceinline__ v8f wmma4(v2f a, v2f b, v8f c) {
  // D = A(16x4,f32) * B(4x16,f32) + C(16x16,f32)
  return __builtin_amdgcn_wmma_f32_16x16x4_f32(
      /*neg_a=*/false, a, /*neg_b=*/false, b,
      /*c_mod=*/(short)0, c, /*reuse_a=*/false, /*reuse_b=*/false);
}

// ---------------------------------------------------------------------------
// Kernel 0: build packed DFT table and zero the accumulator.
//   Epk[p*128 + c] = {Er[2p][c], Er[2p+1][c], Ei[2p][c], Ei[2p+1][c]}
// so one b128 load supplies a full 2-row (k, k+1) B/A fragment pair.
// ---------------------------------------------------------------------------
__global__ __launch_bounds__(256) void donn_init(float4* __restrict__ Epk,
                                                 float* __restrict__ S) {
  int idx = blockIdx.x * 256 + threadIdx.x;
  if (idx < NPIX) S[idx] = 0.0f;
  if (idx < NPIX / 2) {
    int p = idx >> 7, c = idx & 127;
    const float step = 0.049087385212340517f;  // 2*pi/128
    float s0, c0, s1, c1;
    // exp(2*pi*i*k*c/128) == exp(2*pi*i*((k*c) mod 128)/128): keep angle small
    sincosf((float)(((2 * p) * c) & 127) * step, &s0, &c0);
    sincosf((float)(((2 * p + 1) * c) & 127) * step, &s1, &c1);
    Epk[idx] = make_float4(c0, c1, s0, s1);
  }
}

// ---------------------------------------------------------------------------
// Kernel 1: per screen m, T = E * F_m * E via FP32 WMMA; S += |T|^2.
// One workgroup (8 wave32) per m.
//   Stage:  Fs (LDS) = (c0 + i*c1) * sqrt(p_v + 1e-12)   [exp/sqrt once/elem]
//   Per 16-wide column block jb:
//     step A: P(128x16) = Fs * E[:, jb]    (A from LDS, B from packed Epk)
//     step B: Tblk      = E * P            (A from Epk via symmetry, B from LDS)
// ---------------------------------------------------------------------------
__global__ __launch_bounds__(256) void donn_screens(
    const float* __restrict__ c_noise,
    const float4* __restrict__ Epk,
    float* __restrict__ S) {
  __shared__ float2 Fs[NN * FSTR];  // {Fr,Fi}  (~129 KB)
  __shared__ float2 Ps[NN * PSTR];  // {Pr,Pi}  (~17 KB)

  const int m = blockIdx.x;
  const float2* __restrict__ cn2 =
      (const float2*)(c_noise + (size_t)m * NPIX * 2);  // (re,im) pairs

  const int tid  = threadIdx.x;
  const int wave = tid >> 5;
  const int lane = tid & 31;
  const int lm   = lane & 15;   // row/col within 16x16 tile
  const int lh   = lane >> 4;   // half-wave selector (K pairing per ISA layout)
  const int i0   = wave * 16;   // this wave's row tile

  // ---- Stage F into LDS (amp applied once; coalesced b64 reads) ----
  {
    const float INV_NDX  = 976.5625f;                 // 1/(N*DX)
    const float PV_COEF  = 6.283185307179586f * 4e-8f;  // 2*pi*L^2
    const float EXP_COEF = -7.8956835208714864e-7f;     // -2*pi^2*L^2
    for (int idx = tid; idx < NPIX; idx += 256) {
      int k1 = idx >> 7, k2 = idx & 127;
      float f1 = (float)(k1 < 64 ? k1 : k1 - 128) * INV_NDX;
      float f2 = (float)(k2 < 64 ? k2 : k2 - 128) * INV_NDX;
      float amp = sqrtf(PV_COEF * expf(EXP_COEF * (f1 * f1 + f2 * f2)) + 1e-12f);
      float2 c = cn2[idx];
      Fs[k1 * FSTR + k2] = make_float2(c.x * amp, c.y * amp);
    }
  }
  __syncthreads();

  for (int jb = 0; jb < 8; ++jb) {
    const int j0 = jb * 16;

    // ---------------- Step A: P = F * E[:, j0:j0+16] ----------------
    v8f pr = {0, 0, 0, 0, 0, 0, 0, 0};
    v8f qi = {0, 0, 0, 0, 0, 0, 0, 0};
    for (int kk = 0; kk < NN; kk += 4) {
      const int ka = kk + lh * 2;                    // A frag covers K = ka, ka+1
      float2 fa = Fs[(i0 + lm) * FSTR + ka];
      float2 fb = Fs[(i0 + lm) * FSTR + ka + 1];
      v2f ar, ai;
      ar.x = fa.x;  ar.y = fb.x;
      ai.x = fa.y;  ai.y = fb.y;
      v2f ain = -ai;                                 // F32 WMMA has no A/B neg

      float4 e = Epk[(ka >> 1) * NN + j0 + lm];      // rows ka,ka+1 @ col j0+lm
      v2f br, bi;
      br.x = e.x;  br.y = e.y;
      bi.x = e.z;  bi.y = e.w;

      pr = wmma4(ar,  br, pr);   // Re += Fr*Er
      pr = wmma4(ain, bi, pr);   // Re -= Fi*Ei
      qi = wmma4(ar,  bi, qi);   // Im += Fr*Ei
      qi = wmma4(ai,  br, qi);   // Im += Fi*Er
    }
#pragma unroll
    for (int v = 0; v < 8; ++v) {                    // C/D layout: row = v + 8*lh
      int row = i0 + v + lh * 8;
      Ps[row * PSTR + lm] = make_float2(pr[v], qi[v]);
    }
    __syncthreads();                                 // P complete for all rows

    // ---------------- Step B: Tblk = E * P ----------------
    v8f tr = {0, 0, 0, 0, 0, 0, 0, 0};
    v8f ti = {0, 0, 0, 0, 0, 0, 0, 0};
    for (int kk = 0; kk < NN; kk += 4) {
      const int ka = kk + lh * 2;
      // E symmetric: E[i0+lm][ka] == E[ka][i0+lm] -> same packed table, coalesced
      float4 e = Epk[(ka >> 1) * NN + i0 + lm];
      v2f ar, ai;
      ar.x = e.x;  ar.y = e.y;
      ai.x = e.z;  ai.y = e.w;
      v2f ain = -ai;

      float2 p0 = Ps[ka * PSTR + lm];
      float2 p1 = Ps[(ka + 1) * PSTR + lm];
      v2f br, bi;
      br.x = p0.x;  br.y = p1.x;
      bi.x = p0.y;  bi.y = p1.y;

      tr = wmma4(ar,  br, tr);
      tr = wmma4(ain, bi, tr);
      ti = wmma4(ar,  bi, ti);
      ti = wmma4(ai,  br, ti);
    }
#pragma unroll
    for (int v = 0; v < 8; ++v) {
      int row = i0 + v + lh * 8;
      int col = j0 + lm;
      float val = tr[v] * tr[v] + ti[v] * ti[v];     // |T|^2
      atomicAdd(&S[row * NN + col], val);            // global_atomic_add_f32
    }
    __syncthreads();                                 // before P is overwritten
  }
}

// ---------------------------------------------------------------------------
// Kernel 2: out[b] = (1/M) * sum_n x[b,n]^2 * S[fftshift(n)]
// ---------------------------------------------------------------------------
__global__ __launch_bounds__(256) void donn_out(const float* __restrict__ x_in,
                                                const float* __restrict__ S,
                                                float* __restrict__ out,
                                                int Mnum) {
  __shared__ float red[256];
  const int b = blockIdx.x;
  const float* x = x_in + (size_t)b * NPIX;
  float acc = 0.0f;
  for (int idx = threadIdx.x; idx < NPIX; idx += 256) {
    int i = idx >> 7, j = idx & 127;
    float xv = x[idx];
    float s  = S[(((i + 64) & 127) << 7) + ((j + 64) & 127)];
    acc = fmaf(xv * xv, s, acc);
  }
  red[threadIdx.x] = acc;
  __syncthreads();
  for (int off = 128; off > 0; off >>= 1) {
    if (threadIdx.x < off) red[threadIdx.x] += red[threadIdx.x + off];
    __syncthreads();
  }
  if (threadIdx.x == 0) out[b] = red[0] / (float)Mnum;
}

// ---------------------------------------------------------------------------
extern "C" void kernel_launch(void* const* d_in, const int* in_sizes, int n_in,
                              void* d_out, int out_size, void* d_ws, size_t ws_size,
                              hipStream_t stream) {
  const float* x_in    = (const float*)d_in[0];
  // d_in[1] (phases) is analytically eliminated: unit-modulus masks and
  // unitary angular-spectrum propagations preserve the L2 norm the reference
  // finally reduces to, so they cannot affect the output.
  const float* c_noise = (const float*)d_in[2];

  const int M = in_sizes[2] / (NPIX * 2);  // c_noise is (M,128,128,2)
  const int B = in_sizes[0] / NPIX;        // x_in is (B,128,128)

  float4* Epk = (float4*)d_ws;                 // 128 KB packed DFT table
  float*  S   = (float*)d_ws + 4 * (NPIX / 2); // 64 KB accumulator after Epk
  (void)ws_size; (void)n_in; (void)out_size;

  donn_init<<<NPIX / 256, 256, 0, stream>>>(Epk, S);
  donn_screens<<<M, 256, 0, stream>>>(c_noise, Epk, S);
  donn_out<<<B, 256, 0, stream>>>(x_in, S, (float*)d_out, M);
}